// NumpyContains_61933428415675
// MI455X (gfx1250) — compile-verified
//
#include <hip/hip_runtime.h>
#include <hip/hip_bf16.h>
#include <stdint.h>

// Sentinel marking an empty hash slot. INT_MIN cannot silently collide:
// queries equal to the sentinel are tracked via a separate LDS flag.
#define SENT (int)0x80000000

#if defined(__AMDGCN__) && __has_builtin(__builtin_amdgcn_global_load_async_to_lds_b32)
#define USE_ASYNC_LDS 1
typedef __attribute__((address_space(1))) int* as1_int_ptr;
typedef __attribute__((address_space(3))) int* as3_int_ptr;
#else
#define USE_ASYNC_LDS 0
#endif

// Main kernel FIRST in the file so the disasm snippet shows it.
__global__ __launch_bounds__(256) void contains_kernel(
        const int* __restrict__ mainp, const int* __restrict__ queryp,
        int M, int Q, int tsize, int tshift, float* __restrict__ out)
{
    extern __shared__ int smem[];
    int* table = smem;          // tsize entries (open addressing)
    int* qbuf  = smem + tsize;  // Q entries (query staged in LDS)
    __shared__ int hasSent;

    const int tid  = threadIdx.x;
    const int nthr = blockDim.x;
    const unsigned mask = (unsigned)tsize - 1u;

    // --- Phase 1a: kick off the async global->LDS query copy FIRST so the
    // DMA overlaps with the LDS table clear below (ASYNCcnt tracks it).
#if USE_ASYNC_LDS
    for (int i = tid; i < Q; i += nthr) {
        __builtin_amdgcn_global_load_async_to_lds_b32(
            (as1_int_ptr)(queryp + i),
            (as3_int_ptr)(qbuf + i),
            0, 0);
    }
#endif

    // --- Phase 1b: clear the table while the async copy is in flight.
    for (int i = tid; i < tsize; i += nthr) table[i] = SENT;
    if (tid == 0) hasSent = 0;

#if USE_ASYNC_LDS
    // Each wave drains its own ASYNCcnt, then the workgroup barrier makes
    // every wave's staged data visible to all.
    asm volatile("s_wait_asynccnt 0" ::: "memory");
#else
    for (int i = tid; i < Q; i += nthr) qbuf[i] = queryp[i];
#endif
    __syncthreads();

    // --- Phase 2: insert query values (ds_cmpstore_rtn_b32 on LDS)
    for (int i = tid; i < Q; i += nthr) {
        int q = qbuf[i];
        if (q == SENT) { hasSent = 1; continue; }
        unsigned h = ((unsigned)q * 2654435761u) >> tshift;
        for (;;) {
            int old = atomicCAS(&table[h], SENT, q);
            if (old == SENT || old == q) break;
            h = (h + 1u) & mask;
        }
    }
    __syncthreads();

    const int sentFlag = hasSent;
    bool found = false;

    auto probe = [&](int v) {
        if (v == SENT) { found |= (sentFlag != 0); return; }
        unsigned h = ((unsigned)v * 2654435761u) >> tshift;
        for (int p = 0; p < tsize; ++p) {
            int t = table[h];
            if (t == SENT) return;       // definitively absent
            if (t == v) { found = true; return; }
            h = (h + 1u) & mask;
        }
    };

    // --- Phase 3: one streaming pass over main with 128-bit loads
    const int gtid    = blockIdx.x * nthr + tid;
    const int gstride = gridDim.x * nthr;
    const int n4 = M >> 2;
    const int4* m4 = reinterpret_cast<const int4*>(mainp);
    for (int i = gtid; i < n4; i += gstride) {
        __builtin_prefetch(&m4[i + gstride], 0, 0);  // global_prefetch_b8 (speculative, OOB-safe)
        int4 v = m4[i];
        probe(v.x); probe(v.y); probe(v.z); probe(v.w);
    }
    for (int i = (n4 << 2) + gtid; i < M; i += gstride) probe(mainp[i]);

    if (found) out[0] = 1.0f;  // benign same-value race across finders
}

__global__ void contains_init(float* out) {
    if (threadIdx.x == 0 && blockIdx.x == 0) out[0] = 0.0f;
}

extern "C" void kernel_launch(void* const* d_in, const int* in_sizes, int n_in,
                              void* d_out, int out_size, void* d_ws, size_t ws_size,
                              hipStream_t stream) {
    const int* mainp  = (const int*)d_in[0];
    const int* queryp = (const int*)d_in[1];
    const int M = in_sizes[0];
    const int Q = in_sizes[1];
    float* out = (float*)d_out;

    // Table size: next pow2 >= 2*Q (load factor <= 0.5). Q=2048 -> 4096 slots.
    int tsize = 256;
    while (tsize < 2 * Q) tsize <<= 1;
    int lg = 0;
    while ((1 << lg) < tsize) ++lg;
    const int tshift = 32 - lg;

    contains_init<<<1, 64, 0, stream>>>(out);

    const int threads = 256;
    const int blocks  = 128;
    const size_t shmem = (size_t)(tsize + (Q > 0 ? Q : 1)) * sizeof(int);
    contains_kernel<<<blocks, threads, shmem, stream>>>(
        mainp, queryp, M, Q, tsize, tshift, out);
}